// LPNormKernel_48490180771967
// MI455X (gfx1250) — compile-verified
//
#include <hip/hip_runtime.h>
#include <stdint.h>

// L1 (p=1) cdist: out[m,o] = sum_c |x[m,c] - w[c,o]| + b[o]
// M = 8192 rows, C = 288, O = 64. VALU-bound (L1 is not bilinear -> no WMMA
// mapping exists). CDNA5 path used: TDM tensor_load_to_lds stages w (72KB)
// into LDS once per block; inner loop reads w via broadcast ds_load_b128.

#define C_DIM 288
#define O_DIM 64
#define TPB   256          // 8 wave32s per block
#define OPT   8            // outputs per thread
#define OBLK  (O_DIM / OPT)       // 8 output blocks
#define ROWS  (TPB / OBLK)        // 32 rows per block

typedef uint32_t v4u __attribute__((ext_vector_type(4)));
typedef int      v8i __attribute__((ext_vector_type(8)));
typedef int      v4i __attribute__((ext_vector_type(4)));

__global__ __launch_bounds__(TPB) void l1cdist_kernel(
    const float* __restrict__ x, const float* __restrict__ w,
    const float* __restrict__ b, float* __restrict__ out, int M)
{
    __shared__ float wsm[C_DIM * O_DIM];   // 73,728 B of 320 KB WGP LDS

    // ---- Stage w[288][64] into LDS ----------------------------------------
#if defined(__HIP_DEVICE_COMPILE__) && __has_builtin(__builtin_amdgcn_tensor_load_to_lds)
    if (threadIdx.x < 32) {                // wave 0 issues one TDM DMA
        const uint64_t ga = (uint64_t)(uintptr_t)w;
        const uint32_t lds_off = (uint32_t)(uintptr_t)wsm; // low 32b = LDS offset
        const uint32_t NE = (uint32_t)(C_DIM * O_DIM);     // 18432 f32 elements

        v4u g0;
        g0[0] = 1u;                                   // count=1, valid user D#
        g0[1] = lds_off;                              // lds_addr
        g0[2] = (uint32_t)(ga & 0xFFFFFFFFu);         // global_addr[31:0]
        g0[3] = (uint32_t)((ga >> 32) & 0x01FFFFFFu)  // global_addr[56:32]
              | (2u << 30);                           // type = 2 ("image")

        v8i g1;
        g1[0] = 0x00020000;                           // data_size = 4B
        g1[1] = (int)((NE & 0xFFFFu) << 16);          // tensor_dim0[15:0]
        g1[2] = (int)(((NE >> 16) & 0xFFFFu) | (1u << 16)); // dim0 hi | tensor_dim1=1
        g1[3] = (int)((NE & 0xFFFFu) << 16);          // tile_dim0 = 18432
        g1[4] = 1;                                    // tile_dim1 = 1, tile_dim2 = 0
        g1[5] = (int)NE;                              // tensor_dim0_stride
        g1[6] = 0;
        g1[7] = 0;

        v4i zz4 = {0, 0, 0, 0};                       // groups 2/3 unused (2D)
        v8i zz8 = {0, 0, 0, 0, 0, 0, 0, 0};           // trailing group (clang-23 form)
        // 6-arg form: (g0, g1, g2, g3, extra_group, cpol)
        __builtin_amdgcn_tensor_load_to_lds(g0, g1, zz4, zz4, zz8, 0);
        __builtin_amdgcn_s_wait_tensorcnt(0);
    }
#else
    for (int i = threadIdx.x; i < C_DIM * O_DIM; i += TPB) wsm[i] = w[i];
#endif
    __syncthreads();

    // ---- Register-tiled L1 accumulation -----------------------------------
    const int obk = threadIdx.x & (OBLK - 1);         // 0..7  -> 8-col slice
    const int r   = threadIdx.x >> 3;                 // 0..31 -> row in block
    const int row = blockIdx.x * ROWS + r;
    if (row >= M) return;                             // M % ROWS == 0 in practice

    const float* __restrict__ xrow = x + (size_t)row * C_DIM;
    const float* __restrict__ wcol = wsm + obk * OPT;

    float acc[OPT];
#pragma unroll
    for (int j = 0; j < OPT; ++j) acc[j] = 0.0f;

#pragma unroll 2
    for (int c = 0; c < C_DIM; c += 4) {
        const float4 xv = *reinterpret_cast<const float4*>(xrow + c); // 16B aligned
        const float xs[4] = {xv.x, xv.y, xv.z, xv.w};
#pragma unroll
        for (int cc = 0; cc < 4; ++cc) {
            const float v = xs[cc];
            const float* wr = wcol + (c + cc) * O_DIM;
#pragma unroll
            for (int j = 0; j < OPT; j += 4) {
                const float4 wv = *reinterpret_cast<const float4*>(wr + j); // ds_load_b128
                acc[j + 0] += fabsf(v - wv.x);   // v_sub + v_add |d| (abs modifier)
                acc[j + 1] += fabsf(v - wv.y);
                acc[j + 2] += fabsf(v - wv.z);
                acc[j + 3] += fabsf(v - wv.w);
            }
        }
    }

    // ---- Bias + store ------------------------------------------------------
    float* __restrict__ orow = out + (size_t)row * O_DIM + obk * OPT;
    const float* __restrict__ brow = b + obk * OPT;
#pragma unroll
    for (int j = 0; j < OPT; j += 4) {
        float4 r4;
        r4.x = acc[j + 0] + brow[j + 0];
        r4.y = acc[j + 1] + brow[j + 1];
        r4.z = acc[j + 2] + brow[j + 2];
        r4.w = acc[j + 3] + brow[j + 3];
        *reinterpret_cast<float4*>(orow + j) = r4;
    }
}

extern "C" void kernel_launch(void* const* d_in, const int* in_sizes, int n_in,
                              void* d_out, int out_size, void* d_ws, size_t ws_size,
                              hipStream_t stream) {
    const float* x = (const float*)d_in[0];   // [8,32,32,288] f32
    const float* w = (const float*)d_in[1];   // [288,64]      f32
    const float* b = (const float*)d_in[2];   // [64]          f32
    float* out = (float*)d_out;               // [8,1024,64]   f32

    const int M = in_sizes[0] / C_DIM;        // 8192 rows
    const int blocks = (M + ROWS - 1) / ROWS; // 256 blocks of 256 threads
    l1cdist_kernel<<<dim3(blocks), dim3(TPB), 0, stream>>>(x, w, b, out, M);
}